// model_506806141192
// MI455X (gfx1250) — compile-verified
//
#include <hip/hip_runtime.h>

// ---------------------------------------------------------------------------
// EAM force field, N=4096, cutoff^2 = 36.  Compute-bound (inputs ~410KB, fully
// L2-resident in 192MB L2; HBM at 23.3TB/s is irrelevant).  Distance-matrix
// 16x16 tiles are produced on the matrix pipe with one V_WMMA_F32_16X16X4_F32
// (Gram trick: d2 = |pi|^2 + 1*|pj|^2 + pi.(-2pj)).  Pair math uses fast HW
// rcp/rsq/exp and shares exponentials between f_r and the phi "right" term
// (same beta/lambda/r_e => right = (b/fe)*fr, identical log-derivative):
// 4 v_exp_f32 + 4 pow-20 chains per masked pair instead of 6 + 6.
// ---------------------------------------------------------------------------

#define NATOMS 4096
#define CUT2   36.0f
#define NTILE  256          // 4096 / 16 column tiles
#define NCH    8            // j-chunks (parallelism + deterministic combine)
#define TPC    (NTILE/NCH)  // tiles per chunk

typedef __attribute__((ext_vector_type(2))) float v2f;
typedef __attribute__((ext_vector_type(8))) float v8f;

__device__ __forceinline__ float rcpf(float x) { return __builtin_amdgcn_rcpf(x); }
__device__ __forceinline__ float rsqf(float x) { return __builtin_amdgcn_rsqf(x); }

// One "side" (one atom's params) of the pair interaction, values only:
//   cf = exp(-be*(x-1))/(1+(x-la)^20)  ->  fr = fe*cf, right = b*cf
//   cl = exp(-al*(x-1))/(1+(x-ka)^20)  ->  left = a*cl
//   ph = left - right
struct SideV { float fr, ph; };
__device__ __forceinline__ SideV side_val(float r, float re, float fe, float be, float la,
                                          float a, float b, float al, float ka) {
    float inv_re = rcpf(re);
    float x = r * inv_re, xm1 = x - 1.0f;
    float yl = x - la;
    float yl2 = yl*yl, yl4 = yl2*yl2, yl8 = yl4*yl4, yl16 = yl8*yl8;
    float il = rcpf(1.0f + yl16*yl4);
    float cf = __expf(-be*xm1) * il;
    float yk = x - ka;
    float yk2 = yk*yk, yk4 = yk2*yk2, yk8 = yk4*yk4, yk16 = yk8*yk8;
    float ik = rcpf(1.0f + yk16*yk4);
    float cl = __expf(-al*xm1) * ik;
    SideV s;
    s.fr = fe*cf;
    s.ph = a*cl - b*cf;
    return s;
}

// Values + d/dr.  fr and right share the log-derivative sf; left uses sk.
struct SideVD { float fr, frd, ph, phd; };
__device__ __forceinline__ SideVD side_vd(float r, float re, float fe, float be, float la,
                                          float a, float b, float al, float ka) {
    float inv_re = rcpf(re);
    float x = r * inv_re, xm1 = x - 1.0f;
    float yl = x - la;
    float yl2 = yl*yl, yl4 = yl2*yl2, yl8 = yl4*yl4, yl16 = yl8*yl8;
    float yl19 = yl16*yl2*yl;
    float il = rcpf(1.0f + yl16*yl4);
    float cf = __expf(-be*xm1) * il;
    float sf = (-be - 20.0f*yl19*il) * inv_re;   // (log cf)' wrt r
    float yk = x - ka;
    float yk2 = yk*yk, yk4 = yk2*yk2, yk8 = yk4*yk4, yk16 = yk8*yk8;
    float yk19 = yk16*yk2*yk;
    float ik = rcpf(1.0f + yk16*yk4);
    float cl = __expf(-al*xm1) * ik;
    float sk = (-al - 20.0f*yk19*ik) * inv_re;   // (log cl)' wrt r
    float left = a*cl, right = b*cf;
    SideVD s;
    s.fr  = fe*cf;
    s.frd = s.fr*sf;
    s.ph  = left - right;
    s.phd = left*sk - right*sf;
    return s;
}

// ---------------------------------------------------------------------------
// Pass 1: rho_i, pe_i partial sums.  One wave per 16-row i-block per j-chunk.
// ---------------------------------------------------------------------------
__global__ __launch_bounds__(32)
void k_rho_pe(const float* __restrict__ pos, const float* __restrict__ par,
              float* __restrict__ rho_part, float* __restrict__ pe_part) {
    __shared__ float sPar[16][8];   // i-side pair params {re,fe,al,be,a,b,ka,la}
    __shared__ float sPos[16][3];
    __shared__ float sN[16];

    const int lane  = threadIdx.x;
    const int hw    = lane >> 4;     // 0 -> rows 0..7 / K=0,1 ; 1 -> rows 8..15 / K=2,3
    const int l16   = lane & 15;
    const int ib    = blockIdx.x * 16;
    const int chunk = blockIdx.y;

    if (lane < 16) {
        int i = ib + lane;
        float x = pos[3*i], y = pos[3*i+1], z = pos[3*i+2];
        sPos[lane][0] = x; sPos[lane][1] = y; sPos[lane][2] = z;
        sN[lane] = x*x + y*y + z*z;
        const float* p = par + 22*i;
        sPar[lane][0]=p[0]; sPar[lane][1]=p[1]; sPar[lane][2]=p[4]; sPar[lane][3]=p[5];
        sPar[lane][4]=p[6]; sPar[lane][5]=p[7]; sPar[lane][6]=p[8]; sPar[lane][7]=p[9];
    }
    __syncthreads();

    // A (16x4 f32): row m = ib+l16, cols [x, y, z, 1]; lanes 0-15 hold K=0,1.
    v2f A;
    A.x = hw ? sPos[l16][2] : sPos[l16][0];
    A.y = hw ? 1.0f         : sPos[l16][1];
    // C (16x16 f32): C[v] = |p_i|^2 for row M = v + 8*hw.
    v8f C;
#pragma unroll
    for (int v = 0; v < 8; ++v) C[v] = sN[v + hw*8];

    float rho_a[8] = {0.f,0.f,0.f,0.f,0.f,0.f,0.f,0.f};
    float pe_a [8] = {0.f,0.f,0.f,0.f,0.f,0.f,0.f,0.f};

    for (int tj = chunk*TPC; tj < chunk*TPC + TPC; ++tj) {
        int j = tj*16 + l16;
        float xj = pos[3*j], yj = pos[3*j+1], zj = pos[3*j+2];
        float nj = xj*xj + yj*yj + zj*zj;
        // B (4x16 f32): col n = j, rows [-2x, -2y, -2z, |p_j|^2]; lanes 0-15 K=0,1.
        v2f B;
        B.x = hw ? -2.0f*zj : -2.0f*xj;
        B.y = hw ? nj       : -2.0f*yj;
        v8f D = __builtin_amdgcn_wmma_f32_16x16x4_f32(false, A, false, B,
                                                      (short)0, C, false, false);
        const float* pj = par + 22*j;
        float re1=pj[0], fe1=pj[1], al1=pj[4], be1=pj[5];
        float a1 =pj[6], b1 =pj[7], ka1=pj[8], la1=pj[9];
        if (tj + 1 < chunk*TPC + TPC)
            __builtin_prefetch(pos + 3*(tj*16 + 16 + l16), 0, 3);

#pragma unroll
        for (int v = 0; v < 8; ++v) {
            int   row = v + hw*8;
            int   i   = ib + row;
            float d2  = D[v];
            if ((j != i) && (d2 <= CUT2)) {      // execz-skip handles empty bodies
                float invr = rsqf(d2);
                float r    = d2 * invr;          // sqrt(d2), fast path
                float re0=sPar[row][0], fe0=sPar[row][1], al0=sPar[row][2], be0=sPar[row][3];
                float a0 =sPar[row][4], b0 =sPar[row][5], ka0=sPar[row][6], la0=sPar[row][7];
                SideV s0 = side_val(r, re0, fe0, be0, la0, a0, b0, al0, ka0);
                SideV s1 = side_val(r, re1, fe1, be1, la1, a1, b1, al1, ka1);
                float phi01 = 0.5f*(s1.fr*rcpf(s0.fr)*s0.ph + s0.fr*rcpf(s1.fr)*s1.ph);
                rho_a[v] += s1.fr;
                pe_a [v] += phi01;
            }
        }
    }
    // Row sums: each row M lives on 16 lanes of one half-wave.
#pragma unroll
    for (int v = 0; v < 8; ++v) {
        float rs = rho_a[v], ps = pe_a[v];
        for (int off = 8; off; off >>= 1) {
            rs += __shfl_xor(rs, off, 16);
            ps += __shfl_xor(ps, off, 16);
        }
        if (l16 == 0) {
            int i = ib + v + hw*8;
            rho_part[chunk*NATOMS + i] = rs;
            pe_part [chunk*NATOMS + i] = ps;
        }
    }
}

// ---------------------------------------------------------------------------
// Combine chunk partials (fixed order) + embedding F(rho), F'(rho), energy.
// ---------------------------------------------------------------------------
__global__ void k_embed(const float* __restrict__ rho_part, const float* __restrict__ pe_part,
                        const float* __restrict__ par,
                        float* __restrict__ frho_out, float* __restrict__ pe_out,
                        float* __restrict__ Fp_out, float* __restrict__ e_atom) {
    int i = blockIdx.x*blockDim.x + threadIdx.x;
    if (i >= NATOMS) return;
    float rho = 0.f, pe = 0.f;
    for (int c = 0; c < NCH; ++c) { rho += rho_part[c*NATOMS+i]; pe += pe_part[c*NATOMS+i]; }

    const float* p = par + 22*i;
    float rho_e = p[2],  rho_s = p[3];
    float fn0 = p[10], fn1 = p[11], fn2 = p[12], fn3 = p[13];
    float f0  = p[14], f1  = p[15], f2  = p[16], f3  = p[17];
    float eta = p[18], fe  = p[19], rho_n = p[20], rho_0 = p[21];
    float F, Fp;
    if (rho < rho_n) {
        float invrn = rcpf(rho_n);
        float t = rho*invrn - 1.0f;
        F  = fn0 + t*(fn1 + t*(fn2 + t*fn3));
        Fp = (fn1 + t*(2.0f*fn2 + 3.0f*fn3*t)) * invrn;
    } else if (rho < rho_0) {
        float invre = rcpf(rho_e);
        float t = rho*invre - 1.0f;
        F  = f0 + t*(f1 + t*(f2 + t*f3));
        Fp = (f1 + t*(2.0f*f2 + 3.0f*f3*t)) * invre;
    } else {
        float rs = fmaxf(rho, 1e-12f);
        float x  = rs * rcpf(rho_s);
        float lx = __logf(x);
        float xe = __powf(x, eta);
        F  = fe*(1.0f - eta*lx)*xe;
        Fp = -fe*eta*eta*lx*xe * rcpf(rs);   // d/drho [fe(1-eta ln x)x^eta]
    }
    frho_out[i] = F;
    pe_out[i]   = pe;
    Fp_out[i]   = Fp;
    e_atom[i]   = F + pe;                    // e_total = sum(frho) + sum(pe)
}

// Fixed-structure tree reduction -> deterministic e_total.
__global__ void k_etotal(const float* __restrict__ e_atom, float* __restrict__ e_out) {
    __shared__ float s[256];
    int t = threadIdx.x;
    float acc = 0.f;
    for (int k = 0; k < NATOMS/256; ++k) acc += e_atom[t*(NATOMS/256) + k];
    s[t] = acc; __syncthreads();
    for (int off = 128; off; off >>= 1) { if (t < off) s[t] += s[t+off]; __syncthreads(); }
    if (t == 0) *e_out = s[0];
}

// ---------------------------------------------------------------------------
// Pass 2: forces.  g_ij = F'(rho_i)*fr1'(r) + 0.25*phi01'(r); acc += (g/r)*delta.
// ---------------------------------------------------------------------------
__global__ __launch_bounds__(32)
void k_force(const float* __restrict__ pos, const float* __restrict__ par,
             const float* __restrict__ Fp_arr, float* __restrict__ acc_part) {
    __shared__ float sPar[16][8];
    __shared__ float sPos[16][3];
    __shared__ float sN[16];
    __shared__ float sFp[16];

    const int lane  = threadIdx.x;
    const int hw    = lane >> 4;
    const int l16   = lane & 15;
    const int ib    = blockIdx.x * 16;
    const int chunk = blockIdx.y;

    if (lane < 16) {
        int i = ib + lane;
        float x = pos[3*i], y = pos[3*i+1], z = pos[3*i+2];
        sPos[lane][0] = x; sPos[lane][1] = y; sPos[lane][2] = z;
        sN[lane] = x*x + y*y + z*z;
        sFp[lane] = Fp_arr[i];
        const float* p = par + 22*i;
        sPar[lane][0]=p[0]; sPar[lane][1]=p[1]; sPar[lane][2]=p[4]; sPar[lane][3]=p[5];
        sPar[lane][4]=p[6]; sPar[lane][5]=p[7]; sPar[lane][6]=p[8]; sPar[lane][7]=p[9];
    }
    __syncthreads();

    v2f A;
    A.x = hw ? sPos[l16][2] : sPos[l16][0];
    A.y = hw ? 1.0f         : sPos[l16][1];
    v8f C;
#pragma unroll
    for (int v = 0; v < 8; ++v) C[v] = sN[v + hw*8];

    float ax[8] = {0.f,0.f,0.f,0.f,0.f,0.f,0.f,0.f};
    float ay[8] = {0.f,0.f,0.f,0.f,0.f,0.f,0.f,0.f};
    float az[8] = {0.f,0.f,0.f,0.f,0.f,0.f,0.f,0.f};

    for (int tj = chunk*TPC; tj < chunk*TPC + TPC; ++tj) {
        int j = tj*16 + l16;
        float xj = pos[3*j], yj = pos[3*j+1], zj = pos[3*j+2];
        float nj = xj*xj + yj*yj + zj*zj;
        v2f B;
        B.x = hw ? -2.0f*zj : -2.0f*xj;
        B.y = hw ? nj       : -2.0f*yj;
        v8f D = __builtin_amdgcn_wmma_f32_16x16x4_f32(false, A, false, B,
                                                      (short)0, C, false, false);
        const float* pj = par + 22*j;
        float re1=pj[0], fe1=pj[1], al1=pj[4], be1=pj[5];
        float a1 =pj[6], b1 =pj[7], ka1=pj[8], la1=pj[9];
        if (tj + 1 < chunk*TPC + TPC)
            __builtin_prefetch(pos + 3*(tj*16 + 16 + l16), 0, 3);

#pragma unroll
        for (int v = 0; v < 8; ++v) {
            int   row = v + hw*8;
            int   i   = ib + row;
            float d2  = D[v];
            if ((j != i) && (d2 <= CUT2)) {
                float invr = rsqf(d2);
                float r    = d2 * invr;
                float re0=sPar[row][0], fe0=sPar[row][1], al0=sPar[row][2], be0=sPar[row][3];
                float a0 =sPar[row][4], b0 =sPar[row][5], ka0=sPar[row][6], la0=sPar[row][7];
                SideVD s0 = side_vd(r, re0, fe0, be0, la0, a0, b0, al0, ka0);
                SideVD s1 = side_vd(r, re1, fe1, be1, la1, a1, b1, al1, ka1);
                float inv0 = rcpf(s0.fr), inv1 = rcpf(s1.fr);
                float t01 = s1.fr*inv0,   t10 = s0.fr*inv1;
                float dphi = 0.5f*( s0.ph*(s1.frd - t01*s0.frd)*inv0 + t01*s0.phd
                                  + s1.ph*(s0.frd - t10*s1.frd)*inv1 + t10*s1.phd );
                float g  = sFp[row]*s1.frd + 0.25f*dphi;
                float cc = g * invr;                 // g / r
                ax[v] += cc*(xj - sPos[row][0]);
                ay[v] += cc*(yj - sPos[row][1]);
                az[v] += cc*(zj - sPos[row][2]);
            }
        }
    }
#pragma unroll
    for (int v = 0; v < 8; ++v) {
        float sx = ax[v], sy = ay[v], sz = az[v];
        for (int off = 8; off; off >>= 1) {
            sx += __shfl_xor(sx, off, 16);
            sy += __shfl_xor(sy, off, 16);
            sz += __shfl_xor(sz, off, 16);
        }
        if (l16 == 0) {
            int i = ib + v + hw*8;
            float* dst = acc_part + (size_t)chunk*(3*NATOMS) + 3*i;
            dst[0] = sx; dst[1] = sy; dst[2] = sz;
        }
    }
}

__global__ void k_combine_acc(const float* __restrict__ acc_part,
                              const float* __restrict__ mass,
                              float* __restrict__ out_acc) {
    int idx = blockIdx.x*blockDim.x + threadIdx.x;   // over 3N
    if (idx >= 3*NATOMS) return;
    float s = 0.f;
    for (int c = 0; c < NCH; ++c) s += acc_part[(size_t)c*(3*NATOMS) + idx];
    out_acc[idx] = s * rcpf(mass[idx/3]);
}

// ---------------------------------------------------------------------------
extern "C" void kernel_launch(void* const* d_in, const int* in_sizes, int n_in,
                              void* d_out, int out_size, void* d_ws, size_t ws_size,
                              hipStream_t stream) {
    const float* pos  = (const float*)d_in[0];   // [N,3]
    const float* par  = (const float*)d_in[1];   // [N,22]
    const float* mass = (const float*)d_in[2];   // [N]
    float* out = (float*)d_out;                  // frho N | pe N | acc 3N | e_total 1
    float* ws  = (float*)d_ws;

    float* rho_part = ws;                          // NCH*N
    float* pe_part  = rho_part + NCH*NATOMS;       // NCH*N
    float* Fp       = pe_part  + NCH*NATOMS;       // N
    float* e_atom   = Fp       + NATOMS;           // N
    float* acc_part = e_atom   + NATOMS;           // NCH*3N

    float* out_frho = out;
    float* out_pe   = out + NATOMS;
    float* out_acc  = out + 2*NATOMS;
    float* out_e    = out + 5*NATOMS;

    dim3 gridTiles(NATOMS/16, NCH);
    k_rho_pe <<<gridTiles, 32, 0, stream>>>(pos, par, rho_part, pe_part);
    k_embed  <<<(NATOMS+255)/256, 256, 0, stream>>>(rho_part, pe_part, par,
                                                    out_frho, out_pe, Fp, e_atom);
    k_etotal <<<1, 256, 0, stream>>>(e_atom, out_e);
    k_force  <<<gridTiles, 32, 0, stream>>>(pos, par, Fp, acc_part);
    k_combine_acc<<<(3*NATOMS+255)/256, 256, 0, stream>>>(acc_part, mass, out_acc);
}